// ImplicitPointRendMaskHead_20873541058709
// MI455X (gfx1250) — compile-verified
//
#include <hip/hip_runtime.h>
#include <math.h>

typedef float v2f __attribute__((ext_vector_type(2)));
typedef float v8f __attribute__((ext_vector_type(8)));

#define RR   16
#define HC   128
#define WC   128
#define NC   (HC * WC)      // 16384
#define PP   8192
#define CIN  256
#define CHN  256
#define HF   128
#define WF   128
#define KIN  (CIN + CHN)    // 512

// ---------------------------------------------------------------------------
// Kernel 1: per-instance top-k via full bitonic sort in LDS (order matters:
// the reference's output point order == sorted-descending-uncertainty order).
// Packed key: (~orderable(unc)) << 32 | idx  -> ascending u64 sort gives
// descending value with ascending-index tie-break (top_k semantics).
// ---------------------------------------------------------------------------
__global__ __launch_bounds__(1024) void topk_sort_kernel(
    const float* __restrict__ logits, int* __restrict__ idx_out) {
  extern __shared__ unsigned long long skey[];
  const int r   = blockIdx.x;
  const int tid = threadIdx.x;
  const float* lp = logits + (size_t)r * NC;

  for (int i = tid; i < NC; i += 1024) {
    float unc = -fabsf(lp[i]);
    unsigned u  = __float_as_uint(unc);
    unsigned ob = (u & 0x80000000u) ? ~u : (u | 0x80000000u);  // ascending-orderable
    unsigned key = ~ob;                                        // descending
    skey[i] = ((unsigned long long)key << 32) | (unsigned)i;
  }
  __syncthreads();

  for (int k = 2; k <= NC; k <<= 1) {
    for (int j = k >> 1; j > 0; j >>= 1) {
      for (int i = tid; i < NC; i += 1024) {
        int ixj = i ^ j;
        if (ixj > i) {
          unsigned long long a = skey[i], b = skey[ixj];
          bool up = ((i & k) == 0);
          if ((a > b) == up) { skey[i] = b; skey[ixj] = a; }
        }
      }
      __syncthreads();
    }
  }

  for (int p = tid; p < PP; p += 1024)
    idx_out[r * PP + p] = (int)(unsigned)(skey[p] & 0xffffffffu);
}

// ---------------------------------------------------------------------------
// Kernel 2: fused grid-sample + positional encoding + 4-layer MLP.
// Block = 128 threads (4 waves), 32 points per block.
// LDS: X tile [32][516] f32, ping-pong H tiles [32][260] f32, point meta.
// GEMMs use V_WMMA_F32_16X16X4_F32 (full f32 precision, matches reference).
// Wave w: row-block (w&1) (16 points), N-half (w>>1) (8 tiles of 16 cols).
// ---------------------------------------------------------------------------
#define TM   32
#define XLD  516      // 512 + 4 pad -> conflict-free A-fragment reads
#define HLD  260      // 256 + 4 pad

__device__ __forceinline__ void wmma_layer(
    const float* __restrict__ in, int ld, int kd,
    const float* __restrict__ W, const float* __restrict__ bias,
    float* __restrict__ ot, int mblk, int nhalf, int lm, int lk) {
  // A fragment: row m = mblk*16 + lm ; cols k + 2*lk + {0,1}
  const float* arow = in + (mblk * 16 + lm) * ld + 2 * lk;
  for (int nt = nhalf * 8; nt < nhalf * 8 + 8; ++nt) {
    const int ncol = nt * 16 + lm;               // B col n = lane&15 in tile
    const float* wp = W + 2 * lk * CHN + ncol;   // rows k + 2*lk + {0,1}
    v8f acc = {};
#pragma unroll 4
    for (int k = 0; k < kd; k += 4) {
      v2f a; a.x = arow[k];        a.y = arow[k + 1];
      v2f b; b.x = wp[k * CHN];    b.y = wp[k * CHN + CHN];
      acc = __builtin_amdgcn_wmma_f32_16x16x4_f32(
          false, a, false, b, (short)0, acc, false, false);
    }
    const float bn = bias[ncol];
#pragma unroll
    for (int v = 0; v < 8; ++v) {
      float d = acc[v] + bn;                     // D(m,n): m = v + 8*lk
      d = d > 0.0f ? d : 0.0f;                   // ReLU
      ot[(mblk * 16 + v + 8 * lk) * HLD + ncol] = d;
    }
  }
}

__global__ __launch_bounds__(128) void fused_pointhead_kernel(
    const float* __restrict__ features, const float* __restrict__ pe_gauss,
    const float* __restrict__ W0, const float* __restrict__ b0,
    const float* __restrict__ W1, const float* __restrict__ b1,
    const float* __restrict__ W2, const float* __restrict__ b2,
    const float* __restrict__ W3, const float* __restrict__ b3,
    const int* __restrict__ idx_in, float* __restrict__ out) {
  extern __shared__ float smem[];
  float* bufX  = smem;                    // [TM][XLD]
  float* bufH0 = bufX + TM * XLD;         // [TM][HLD]
  float* bufH1 = bufH0 + TM * HLD;        // [TM][HLD]
  float* m_px  = bufH1 + TM * HLD;
  float* m_py  = m_px + TM;
  float* m_wx  = m_py + TM;
  float* m_wy  = m_wx + TM;
  int*   m_x0  = (int*)(m_wy + TM);
  int*   m_y0  = m_x0 + TM;

  const int r     = blockIdx.y;
  const int pbase = blockIdx.x * TM;
  const int tid   = threadIdx.x;

  if (tid == 0) {  // pull weights toward L2/WGP$ (global_prefetch_b8)
    __builtin_prefetch(W0, 0, 3);
    __builtin_prefetch(W1, 0, 3);
    __builtin_prefetch(W2, 0, 3);
  }

  // ---- stage 1: per-point coordinates / bilinear meta ----
  if (tid < TM) {
    const int idx = idx_in[r * PP + pbase + tid];
    const float px = 0.5f / WC + (float)(idx & (WC - 1)) * (1.0f / WC);
    const float py = 0.5f / HC + (float)(idx >> 7) * (1.0f / HC);
    const float fx = (WF / 0.25f - 2.0f) / (WF / 0.25f - 1.0f);  // 510/511
    const float fy = (HF / 0.25f - 2.0f) / (HF / 0.25f - 1.0f);
    const float gx = 2.0f * px * fx - 1.0f;
    const float gy = 2.0f * py * fy - 1.0f;
    const float x = ((gx + 1.0f) * WF - 1.0f) * 0.5f;
    const float y = ((gy + 1.0f) * HF - 1.0f) * 0.5f;
    const float x0f = floorf(x), y0f = floorf(y);
    m_px[tid] = px;       m_py[tid] = py;
    m_wx[tid] = x - x0f;  m_wy[tid] = y - y0f;
    m_x0[tid] = (int)x0f; m_y0[tid] = (int)y0f;
  }
  __syncthreads();

  // ---- stage 2: bilinear grid-sample (zero padding) -> X[:, 256..511] ----
  for (int e = tid; e < TM * CIN; e += 128) {
    const int m = e >> 8;
    const int c = e & (CIN - 1);
    const int x0 = m_x0[m], y0 = m_y0[m];
    const float wx1 = m_wx[m], wy1 = m_wy[m];
    const float wx0 = 1.0f - wx1, wy0 = 1.0f - wy1;
    const float* fb = features + ((size_t)r * CIN + c) * (HF * WF);
    const bool vx0 = (x0 >= 0) && (x0 < WF);
    const bool vx1 = (x0 + 1 >= 0) && (x0 + 1 < WF);
    const bool vy0 = (y0 >= 0) && (y0 < HF);
    const bool vy1 = (y0 + 1 >= 0) && (y0 + 1 < HF);
    float v = 0.0f;
    if (vy0 && vx0) v += wx0 * wy0 * fb[y0 * WF + x0];
    if (vy0 && vx1) v += wx1 * wy0 * fb[y0 * WF + x0 + 1];
    if (vy1 && vx0) v += wx0 * wy1 * fb[(y0 + 1) * WF + x0];
    if (vy1 && vx1) v += wx1 * wy1 * fb[(y0 + 1) * WF + x0 + 1];
    bufX[m * XLD + CHN + c] = v;
  }

  // ---- stage 3: Fourier positional encoding -> X[:, 0..255] ----
  for (int e = tid; e < TM * (CHN / 2); e += 128) {
    const int m = e >> 7;
    const int c = e & (CHN / 2 - 1);
    const float lx = 2.0f * m_px[m] - 1.0f;
    const float ly = 2.0f * m_py[m] - 1.0f;
    const float proj =
        6.28318530717958647692f * (lx * pe_gauss[c] + ly * pe_gauss[CHN / 2 + c]);
    bufX[m * XLD + c] = __sinf(proj);
    bufX[m * XLD + CHN / 2 + c] = __cosf(proj);
  }
  __syncthreads();

  // ---- WMMA MLP: 512->256->256->256 with ReLU ----
  const int wave  = tid >> 5;
  const int lane  = tid & 31;
  const int mblk  = wave & 1;
  const int nhalf = wave >> 1;
  const int lm    = lane & 15;
  const int lk    = lane >> 4;

  wmma_layer(bufX,  XLD, KIN, W0, b0, bufH0, mblk, nhalf, lm, lk);
  __syncthreads();
  wmma_layer(bufH0, HLD, CHN, W1, b1, bufH1, mblk, nhalf, lm, lk);
  __syncthreads();
  wmma_layer(bufH1, HLD, CHN, W2, b2, bufH0, mblk, nhalf, lm, lk);
  __syncthreads();

  // ---- final 256 -> 1 projection ----
  if (tid < TM) {
    const float* h = bufH0 + tid * HLD;
    float s = 0.0f;
#pragma unroll 8
    for (int c = 0; c < CHN; ++c) s += h[c] * W3[c];
    out[(size_t)r * PP + pbase + tid] = s + b3[0];
  }
}

// ---------------------------------------------------------------------------
extern "C" void kernel_launch(void* const* d_in, const int* in_sizes, int n_in,
                              void* d_out, int out_size, void* d_ws, size_t ws_size,
                              hipStream_t stream) {
  const float* coarse   = (const float*)d_in[0];
  const float* features = (const float*)d_in[1];
  const float* pe_gauss = (const float*)d_in[2];
  const float* W0 = (const float*)d_in[3];
  const float* b0 = (const float*)d_in[4];
  const float* W1 = (const float*)d_in[5];
  const float* b1 = (const float*)d_in[6];
  const float* W2 = (const float*)d_in[7];
  const float* b2 = (const float*)d_in[8];
  const float* W3 = (const float*)d_in[9];
  const float* b3 = (const float*)d_in[10];
  float* out = (float*)d_out;
  int* idxbuf = (int*)d_ws;  // 16 * 8192 int32 = 512 KB

  // Kernel 1: per-instance bitonic top-k sort (128 KB LDS each).
  topk_sort_kernel<<<RR, 1024, NC * sizeof(unsigned long long), stream>>>(
      coarse, idxbuf);

  // Kernel 2: fused grid-sample + PE + WMMA MLP.
  const size_t smem2 =
      (size_t)(TM * XLD + 2 * TM * HLD + 4 * TM) * sizeof(float) +
      (size_t)(2 * TM) * sizeof(int);  // = 133,376 B (fits 320 KB WGP LDS)
  dim3 grid(PP / TM, RR);
  fused_pointhead_kernel<<<grid, 128, smem2, stream>>>(
      features, pe_gauss, W0, b0, W1, b1, W2, b2, W3, b3, idxbuf, out);
}